// MinibatchDiscrimination_21303037788215
// MI455X (gfx1250) — compile-verified
//
#include <hip/hip_runtime.h>

typedef float v2f __attribute__((ext_vector_type(2)));
typedef float v8f __attribute__((ext_vector_type(8)));

// ---------------------------------------------------------------------------
// Stage 1:  M(1024x256) = x(1024x256) @ W(256x256),
//           where W[i][o*8+l] = kernel[o][i][l]   (kernel is (32,256,8))
// One wave computes one 16x16 tile of M using V_WMMA_F32_16X16X4_F32.
// 8 waves / block, 128 blocks -> 1024 tiles.
// ---------------------------------------------------------------------------
__global__ __launch_bounds__(256) void mbd_gemm_f32wmma(
    const float* __restrict__ x,
    const float* __restrict__ kern,
    float* __restrict__ M)
{
    const int lane   = threadIdx.x & 31;
    const int wave   = threadIdx.x >> 5;
    const int waveId = blockIdx.x * 8 + wave;   // 0..1023
    const int tm     = waveId >> 4;             // 0..63  (row tile)
    const int tn     = waveId & 15;             // 0..15  (col tile)

    const int l16  = lane & 15;                 // M (for A) / N (for B,C,D)
    const int half = lane >> 4;                 // K-group selector
    const int r0   = tm * 16;
    const int c0   = tn * 16;

    // A-matrix f32 16x4 layout: lane holds row M=l16; VGPR v -> K = 2*half + v.
    // Elements (k+2*half, k+2*half+1) are contiguous -> one float2 load.
    const float* arow = x + (r0 + l16) * 256 + 2 * half;

    // B-matrix f32 4x16 layout: lane holds col N=l16; VGPR v -> K = 2*half + v.
    // B row i, col j lives at kernel[(j>>3)*2048 + i*8 + (j&7)].
    const int    j    = c0 + l16;
    const float* wcol = kern + (j >> 3) * 2048 + (j & 7);
    const int    kb   = 2 * half;

    v8f acc = {};
#pragma unroll 4
    for (int k = 0; k < 256; k += 4) {
        v2f a = *(const v2f*)(arow + k);
        v2f b;
        b.x = wcol[(k + kb) * 8];
        b.y = wcol[(k + kb + 1) * 8];
        // (neg_a, A, neg_b, B, c_mod, C, reuse_a, reuse_b)
        acc = __builtin_amdgcn_wmma_f32_16x16x4_f32(
                  false, a, false, b, (short)0, acc, false, false);
    }

    // C/D layout: VGPR r -> row M = r + 8*half, col N = l16.
#pragma unroll
    for (int r = 0; r < 8; ++r) {
        M[(r0 + r + 8 * half) * 256 + c0 + l16] = acc[r];
    }
}

// ---------------------------------------------------------------------------
// Stage 2: for each o, pairwise L1 over L=8 dims, out[b,o] = mean_b' exp(-L1).
// Block = (row tile of 256 b's, one o). Full 1024x8 slice staged in LDS (32KB).
// ---------------------------------------------------------------------------
__global__ __launch_bounds__(256) void mbd_pairwise(
    const float* __restrict__ M,
    float* __restrict__ out)
{
    __shared__ float lds[1024 * 8];             // 32 KB

    const int o   = blockIdx.y;                 // 0..31
    const int tid = threadIdx.x;                // 0..255

    // Stage M[:, o, :] -> LDS as lds[b*8 + l], float4 chunks.
    const float* src = M + o * 8;
    for (int i = tid; i < 2048; i += 256) {
        const int b = i >> 1;
        const int h = i & 1;
        *(float4*)(&lds[i * 4]) = *(const float4*)(src + b * 256 + h * 4);
    }
    __syncthreads();

    const int b = blockIdx.x * 256 + tid;       // this thread's row
    float m[8];
#pragma unroll
    for (int l = 0; l < 8; ++l) m[l] = lds[b * 8 + l];

    const float kLog2e = 1.4426950408889634f;
    float acc = 0.0f;
#pragma unroll 4
    for (int bp = 0; bp < 1024; ++bp) {
        const float* q = &lds[bp * 8];          // wave-uniform -> LDS broadcast
        float l1 = 0.0f;
#pragma unroll
        for (int l = 0; l < 8; ++l) l1 += __builtin_fabsf(m[l] - q[l]);
        acc += __builtin_amdgcn_exp2f(-l1 * kLog2e);   // exp(-l1)
    }
    out[b * 32 + o] = acc * (1.0f / 1024.0f);
}

// ---------------------------------------------------------------------------
extern "C" void kernel_launch(void* const* d_in, const int* in_sizes, int n_in,
                              void* d_out, int out_size, void* d_ws, size_t ws_size,
                              hipStream_t stream)
{
    const float* x    = (const float*)d_in[0];   // (1024, 256) f32
    const float* kern = (const float*)d_in[1];   // (32, 256, 8) f32
    float*       out  = (float*)d_out;           // (1024, 32) f32
    float*       M    = (float*)d_ws;            // (1024, 256) f32 scratch, 1 MB

    mbd_gemm_f32wmma<<<128, 256, 0, stream>>>(x, kern, M);
    mbd_pairwise<<<dim3(4, 32), 256, 0, stream>>>(M, out);
}